// TopKnoGhost_32890859553025
// MI455X (gfx1250) — compile-verified
//
#include <hip/hip_runtime.h>
#include <cmath>

#ifndef M_PI
#define M_PI 3.14159265358979323846
#endif

// ---------------------------------------------------------------------------
// TopKnoGhost forward for x[64,256,32,32] f32.
// One workgroup per channel; whole channel (256KB) staged in LDS via the
// CDNA5 Tensor Data Mover (tensor_load_to_lds + s_wait_tensorcnt), then
// mean / exact radix-select top-100 of |x-mean| / affine output all run out
// of LDS.  HBM traffic = 1 read + 1 write = 128MB (~5.5us floor @ 23.3TB/s).
// Radix levels 2-3 run on a compacted candidate array (integer counting is
// order-independent -> deterministic); all float reductions are fixed-order
// full sweeps (bit-deterministic across replays).
// ---------------------------------------------------------------------------

typedef __attribute__((ext_vector_type(4))) unsigned int u32x4;
typedef __attribute__((ext_vector_type(8))) int          i32x8;
typedef __attribute__((ext_vector_type(4))) int          i32x4;

#define C_CH     256
#define N_B      64
#define HW       1024          // 32*32 contiguous floats per (n, c)
#define M_TOT    65536         // N_B * HW
#define K_TOP    100
#define BLK      1024          // 32 wave32s
#define CAND_CAP 12288         // candidate key array (48KB)
#define SMEM_BYTES (M_TOT * 4 + 256 * 4 + CAND_CAP * 4)   // ~313KB < 320KB/WGP

// Deterministic block-wide sum: fixed shuffle tree per wave, fixed LDS tree
// across the 32 waves.  Result broadcast to all threads.
__device__ __forceinline__ float block_reduce_sum(float v, float* red) {
    const int lane = threadIdx.x & 31;
    const int wv   = (int)(threadIdx.x >> 5);
#pragma unroll
    for (int o = 16; o > 0; o >>= 1) v += __shfl_down(v, o, 32);
    if (lane == 0) red[wv] = v;
    __syncthreads();
    if (wv == 0) {
        float t = red[lane];
#pragma unroll
        for (int o = 16; o > 0; o >>= 1) t += __shfl_down(t, o, 32);
        if (lane == 0) red[0] = t;
    }
    __syncthreads();
    float r = red[0];
    __syncthreads();           // protect red[] reuse by the next reduction
    return r;
}

// Pick the bin holding the `need`-th largest key (counting from the top),
// update prefix/need.  hist[] must be final; result broadcast via LDS.
#define SELECT_BIN(SHIFT)                                                 \
    __syncthreads();                                                      \
    if (tid == 0) {                                                       \
        unsigned int acc = 0u;                                            \
        for (int b = 255; b >= 0; --b) {                                  \
            unsigned int h = hist[b];                                     \
            if (acc + h >= need) {                                        \
                s_prefix = prefix | ((unsigned int)b << (SHIFT));         \
                s_need   = need - acc;                                    \
                break;                                                    \
            }                                                             \
            acc += h;                                                     \
        }                                                                 \
    }                                                                     \
    __syncthreads();                                                      \
    prefix = s_prefix;                                                    \
    need   = s_need;                                                      \
    __syncthreads();

__global__ void __launch_bounds__(BLK)
TopKnoGhost_32890859553025_kernel(const float* __restrict__ x,
                                  const float* __restrict__ weight,
                                  const float* __restrict__ bias,
                                  float* __restrict__ out,
                                  float kconst) {
    extern __shared__ float smem[];
    float*        data = smem;                                // 65536 floats
    unsigned int* hist = (unsigned int*)(smem + M_TOT);       // 256 bins
    unsigned int* cand = hist + 256;                          // CAND_CAP keys
    __shared__ float red[32];
    __shared__ unsigned int s_prefix, s_need, s_ccount;

    const int c   = blockIdx.x;
    const int tid = (int)threadIdx.x;

    // ---- Stage the channel into LDS with the Tensor Data Mover -----------
    // 2-D tile: tile_dim0 = 1024 elements (contiguous H*W), tile_dim1 = 64
    // rows (batch), row stride = C*H*W = 262144 elements.  Wave 0 issues a
    // single descriptor; TENSORcnt tracks completion.
    if ((tid >> 5) == 0) {   // uniform per-wave guard -> exactly one TDM op
        unsigned long long gaddr =
            (unsigned long long)(const void*)(x + (size_t)c * HW);
        unsigned int lds = (unsigned int)(size_t)(void*)data; // low 32b = LDS byte addr

        u32x4 g0 = {
            1u,                                            // count=1, user mode
            lds,                                           // lds_addr
            (unsigned int)(gaddr & 0xffffffffull),         // global_addr[31:0]
            (unsigned int)((gaddr >> 32) & 0x01ffffffull)  // global_addr[56:32]
                | (2u << 30)                               // type = 2 (image)
        };
        i32x8 g1 = {
            (int)(2u << 16),           // data_size = 4B
            (int)((unsigned)HW << 16), // tensor_dim0[15:0]   (=1024)
            (int)((unsigned)N_B << 16),// tensor_dim1[15:0]   (=64)
            (int)((unsigned)HW << 16), // tile_dim0           (=1024)
            (int)N_B,                  // tile_dim1           (=64)
            (int)(C_CH * HW),          // tensor_dim0_stride low 32 (=262144)
            0, 0                       // stride hi / tensor_dim1_stride (2D)
        };
        i32x4 z4 = {0, 0, 0, 0};
#if defined(__clang_major__) && (__clang_major__ >= 23)
        i32x8 z8 = {0, 0, 0, 0, 0, 0, 0, 0};
        __builtin_amdgcn_tensor_load_to_lds(g0, g1, z4, z4, z8, 0);
#else
        __builtin_amdgcn_tensor_load_to_lds(g0, g1, z4, z4, 0);
#endif
    }
    __builtin_amdgcn_s_wait_tensorcnt(0);
    __syncthreads();

    // ---- Sweep 1: per-channel mean (vectorized LDS reads) -----------------
    float part = 0.f;
#pragma unroll
    for (int i = tid * 4; i < M_TOT; i += BLK * 4) {
        float4 v = *(const float4*)(data + i);
        part += (v.x + v.y) + (v.z + v.w);
    }
    const float mean = block_reduce_sum(part, red) * (1.0f / (float)M_TOT);

    // ---- Radix select the K-th largest |x - mean| -------------------------
    // Keys are IEEE bit patterns of non-negative floats => monotonic in value.
    unsigned int prefix = 0u;
    unsigned int need   = K_TOP;

    // Sweep 2 (level 0): full-data histogram on top byte.
    if (tid < 256) hist[tid] = 0u;
    if (tid == 0) s_ccount = 0u;
    __syncthreads();
    for (int i = tid * 4; i < M_TOT; i += BLK * 4) {
        float4 v = *(const float4*)(data + i);
#define H0(val)                                                           \
        atomicAdd(&hist[__float_as_uint(fabsf((val) - mean)) >> 24], 1u);
        H0(v.x) H0(v.y) H0(v.z) H0(v.w)
#undef H0
    }
    SELECT_BIN(24)

    // Sweep 3 (level 1): full sweep; matching elements feed the level-1
    // histogram AND are compacted into cand[] for levels 2-3.
    if (tid < 256) hist[tid] = 0u;
    __syncthreads();
    for (int i = tid * 4; i < M_TOT; i += BLK * 4) {
        float4 v = *(const float4*)(data + i);
#define H1(val)                                                           \
        {                                                                 \
            unsigned int key = __float_as_uint(fabsf((val) - mean));      \
            if ((key & 0xff000000u) == prefix) {                          \
                atomicAdd(&hist[(key >> 16) & 0xffu], 1u);                \
                unsigned int idx = atomicAdd(&s_ccount, 1u);              \
                if (idx < CAND_CAP) cand[idx] = key;                      \
            }                                                             \
        }
        H1(v.x) H1(v.y) H1(v.z) H1(v.w)
#undef H1
    }
    SELECT_BIN(16)

    const unsigned int ccount  = s_ccount;          // written before barrier
    const bool         useCand = (ccount <= CAND_CAP);

    // Levels 2-3: run on the compacted candidates (integer counting is
    // order-independent => deterministic); fall back to full sweeps on
    // overflow (pathological distributions).
    for (int level = 2; level < 4; ++level) {
        const int shift = 24 - 8 * level;
        const unsigned int kmask = 0xffffffffu << (shift + 8);
        if (tid < 256) hist[tid] = 0u;
        __syncthreads();
        if (useCand) {
            for (int i = tid; i < (int)ccount; i += BLK) {
                unsigned int key = cand[i];
                if ((key & kmask) == prefix)
                    atomicAdd(&hist[(key >> shift) & 0xffu], 1u);
            }
        } else {
            for (int i = tid * 4; i < M_TOT; i += BLK * 4) {
                float4 v = *(const float4*)(data + i);
#define HL(val)                                                           \
                {                                                         \
                    unsigned int key =                                    \
                        __float_as_uint(fabsf((val) - mean));             \
                    if ((key & kmask) == prefix)                          \
                        atomicAdd(&hist[(key >> shift) & 0xffu], 1u);     \
                }
                HL(v.x) HL(v.y) HL(v.z) HL(v.w)
#undef HL
            }
        }
        SELECT_BIN(shift)
    }

    // T = exact key of the K-th largest; need = #ties of T inside top-K.
    const unsigned int T = prefix;
    const float tval = __uint_as_float(T);

    // ---- Sweep 4: deterministic tie-aware top-K sum -----------------------
    float psum = 0.f;
#pragma unroll
    for (int i = tid * 4; i < M_TOT; i += BLK * 4) {
        float4 v = *(const float4*)(data + i);
#define SUMP(val)                                                         \
        {                                                                 \
            unsigned int key = __float_as_uint(fabsf((val) - mean));      \
            if (key > T) psum += __uint_as_float(key);                    \
        }
        SUMP(v.x) SUMP(v.y) SUMP(v.z) SUMP(v.w)
#undef SUMP
    }
    const float sum_gt = block_reduce_sum(psum, red);
    const float mean_topk =
        (sum_gt + (float)need * tval) * (1.0f / (float)K_TOP) * kconst;
    const float scale = 1.0f / (mean_topk + 1e-7f);

    // ---- Sweep 5: affine output, float4 coalesced stores ------------------
    const float a  = scale * weight[c];
    const float bb = bias[c];
    const size_t cbase = (size_t)c * HW;
#pragma unroll
    for (int i = tid * 4; i < M_TOT; i += BLK * 4) {
        float4 v = *(const float4*)(data + i);
        const int n   = i >> 10;          // batch row in LDS slab
        const int pos = i & 1023;         // position within H*W
        float4 o;
        o.x = (v.x - mean) * a + bb;
        o.y = (v.y - mean) * a + bb;
        o.z = (v.z - mean) * a + bb;
        o.w = (v.w - mean) * a + bb;
        *(float4*)(out + (size_t)n * C_CH * HW + cbase + (size_t)pos) = o;
    }
}

extern "C" void kernel_launch(void* const* d_in, const int* in_sizes, int n_in,
                              void* d_out, int out_size, void* d_ws, size_t ws_size,
                              hipStream_t stream) {
    const float* x = (const float*)d_in[0];
    const float* w = (const float*)d_in[1];
    const float* b = (const float*)d_in[2];
    float* out = (float*)d_out;

    // const = 0.5*(1+sqrt(pi*ln4))/sqrt(2*ln M), M = 65536
    const float kconst = (float)(0.5 * (1.0 + sqrt(M_PI * log(4.0))) /
                                 sqrt(2.0 * log((double)M_TOT)));

    hipFuncSetAttribute(
        reinterpret_cast<const void*>(TopKnoGhost_32890859553025_kernel),
        hipFuncAttributeMaxDynamicSharedMemorySize, SMEM_BYTES);

    TopKnoGhost_32890859553025_kernel<<<C_CH, BLK, SMEM_BYTES, stream>>>(
        x, w, b, out, kconst);
}